// QFT_9045201125418
// MI455X (gfx1250) — compile-verified
//
#include <hip/hip_runtime.h>

typedef float v2f __attribute__((ext_vector_type(2)));
typedef float v8f __attribute__((ext_vector_type(8)));

#define N_GATES 435
#define BATCH   (1u << 21)
#define TPW     4          // iterations of 32 columns per wave -> 128 cols/wave

// ---------------------------------------------------------------------------
// Kernel 1: fold the 435 complex 4x4 gates into one operator.
// Matrix slot layout: 32 floats = [0..15] real 4x4 row-major, [16..31] imag.
// Order-preserving pairwise tree: combine(i) = M[2i+1] @ M[2i].
// ---------------------------------------------------------------------------
__device__ __forceinline__ void cmat_mul(const float* A, const float* B, float* C) {
#pragma unroll
  for (int r = 0; r < 4; ++r) {
#pragma unroll
    for (int c = 0; c < 4; ++c) {
      float cr = 0.f, ci = 0.f;
#pragma unroll
      for (int k = 0; k < 4; ++k) {
        float ar = A[r * 4 + k], ai = A[16 + r * 4 + k];
        float br = B[k * 4 + c], bi = B[16 + k * 4 + c];
        cr = __builtin_fmaf(ar, br, cr);
        cr = __builtin_fmaf(-ai, bi, cr);
        ci = __builtin_fmaf(ar, bi, ci);
        ci = __builtin_fmaf(ai, br, ci);
      }
      C[r * 4 + c] = cr;
      C[16 + r * 4 + c] = ci;
    }
  }
}

__global__ void fold_gates_kernel(const float* __restrict__ u, float* __restrict__ w) {
  __shared__ float bufA[218 * 32];   // 27,904 B
  __shared__ float bufB[109 * 32];   // 13,952 B  (total 41,856 B LDS)
  const int t = threadIdx.x;

  // level 0: 435 -> 218 (read gates from global)
  {
    const int pairs = N_GATES / 2;   // 217
    if (t < pairs) {
      cmat_mul(u + (2 * t + 1) * 32, u + (2 * t) * 32, bufA + t * 32);
    } else if (t == pairs) {
#pragma unroll
      for (int i = 0; i < 32; ++i) bufA[pairs * 32 + i] = u[(N_GATES - 1) * 32 + i];
    }
  }
  __syncthreads();

  float* src = bufA;
  float* dst = bufB;
  int n = 218;
  while (n > 1) {                    // 218->109->55->28->14->7->4->2->1
    const int pairs = n >> 1;
    if (t < pairs) cmat_mul(src + (2 * t + 1) * 32, src + (2 * t) * 32, dst + t * 32);
    if ((n & 1) && t == pairs) {
      for (int i = 0; i < 32; ++i) dst[pairs * 32 + i] = src[(n - 1) * 32 + i];
    }
    __syncthreads();
    float* tmp = src; src = dst; dst = tmp;
    n = pairs + (n & 1);
  }

  // src[0..31] = P = U_434 @ ... @ U_0.  Expand to real 8x8: W = [[Pr,-Pi],[Pi,Pr]]
  if (t < 64) {
    const int r = t >> 3, c = t & 7;
    const int rr = r & 3, cc = c & 3;
    const float pre = src[rr * 4 + cc];
    const float pim = src[16 + rr * 4 + cc];
    float val;
    if (r < 4) val = (c < 4) ? pre : -pim;
    else       val = (c < 4) ? pim :  pre;
    w[t] = val;
  }
}

// ---------------------------------------------------------------------------
// Kernel 2: streaming apply  out(8 x BATCH) = W(8x8) * S(8 x BATCH)
// via V_WMMA_F32_16X16X4_F32:  D(16x16) = A(16x4) B(4x16) + C, two K=4 chunks.
// A rows 0..7 = W chunk, rows 8..15 = 0.  B columns = 16 batch columns.
// Two tiles per iteration; ds_bpermute splices tile-B rows into lanes 16..31
// so every store is a full-wave 128B contiguous nontemporal store.
// ---------------------------------------------------------------------------
__global__ void apply_kernel(const float* __restrict__ x,
                             const float* __restrict__ w,
                             float* __restrict__ out) {
  const int lane = threadIdx.x & 31;
  const unsigned wid = (unsigned)(blockIdx.x * blockDim.x + threadIdx.x) >> 5;
  const int m  = lane & 15;          // A row / B,D column-in-tile
  const int kb = (lane >> 4) << 1;   // K pair base: lanes 0-15 -> {0,1}, 16-31 -> {2,3}

  // A operands (ISA 16x4 f32 layout: VGPR0=K kb, VGPR1=K kb+1 per lane-half)
  v2f a0 = {0.f, 0.f}, a1 = {0.f, 0.f};
  if (m < 8) {
    a0.x = w[m * 8 + kb];
    a0.y = w[m * 8 + kb + 1];
    a1.x = w[m * 8 + 4 + kb];
    a1.y = w[m * 8 + 4 + kb + 1];
  }

  unsigned base = wid * (32u * TPW);
#pragma unroll
  for (int it = 0; it < TPW; ++it, base += 32u) {
    const unsigned colA = base + (unsigned)m;
    const unsigned colB = colA + 16u;

    // B operands: component c lives at x + c*BATCH (c<4 real rows, c>=4 imag rows)
    v2f b0A, b1A, b0B, b1B;
    b0A.x = __builtin_nontemporal_load(&x[(unsigned)(kb + 0) * BATCH + colA]);
    b0A.y = __builtin_nontemporal_load(&x[(unsigned)(kb + 1) * BATCH + colA]);
    b1A.x = __builtin_nontemporal_load(&x[(unsigned)(kb + 4) * BATCH + colA]);
    b1A.y = __builtin_nontemporal_load(&x[(unsigned)(kb + 5) * BATCH + colA]);
    b0B.x = __builtin_nontemporal_load(&x[(unsigned)(kb + 0) * BATCH + colB]);
    b0B.y = __builtin_nontemporal_load(&x[(unsigned)(kb + 1) * BATCH + colB]);
    b1B.x = __builtin_nontemporal_load(&x[(unsigned)(kb + 4) * BATCH + colB]);
    b1B.y = __builtin_nontemporal_load(&x[(unsigned)(kb + 5) * BATCH + colB]);

    v8f accA = {};
    accA = __builtin_amdgcn_wmma_f32_16x16x4_f32(false, a0, false, b0A,
                                                 (short)0, accA, false, false);
    accA = __builtin_amdgcn_wmma_f32_16x16x4_f32(false, a1, false, b1A,
                                                 (short)0, accA, false, false);
    v8f accB = {};
    accB = __builtin_amdgcn_wmma_f32_16x16x4_f32(false, a0, false, b0B,
                                                 (short)0, accB, false, false);
    accB = __builtin_amdgcn_wmma_f32_16x16x4_f32(false, a1, false, b1B,
                                                 (short)0, accB, false, false);

    // D row r (lanes 0-15 of VGPR r) -> component r.  Merge tileA|tileB and store 32 cols.
#pragma unroll
    for (int r = 0; r < 8; ++r) {
      const int hv = __builtin_amdgcn_ds_bpermute(m << 2,
                        __builtin_bit_cast(int, accB[r]));
      const float val = (lane < 16) ? accA[r] : __builtin_bit_cast(float, hv);
      __builtin_nontemporal_store(val, &out[(unsigned)r * BATCH + base + (unsigned)lane]);
    }
  }
}

// ---------------------------------------------------------------------------
extern "C" void kernel_launch(void* const* d_in, const int* in_sizes, int n_in,
                              void* d_out, int out_size, void* d_ws, size_t ws_size,
                              hipStream_t stream) {
  const float* u  = (const float*)d_in[0];   // [435, 2, 4, 4]
  const float* x  = (const float*)d_in[1];   // [2, 4, BATCH]
  float* out      = (float*)d_out;           // [2, 4, BATCH]
  float* w        = (float*)d_ws;            // 8x8 folded real operator (256 B)

  fold_gates_kernel<<<1, 256, 0, stream>>>(u, w);

  const unsigned waves  = BATCH / (32u * TPW);       // 16384 waves, exact tiling
  const unsigned block  = 256;                       // 8 waves/block
  const unsigned blocks = waves / (block / 32u);     // 2048 blocks
  apply_kernel<<<blocks, block, 0, stream>>>(x, w, out);
}